// MuellerMatrixPyramid_72430328480041
// MI455X (gfx1250) — compile-verified
//
#include <hip/hip_runtime.h>
#include <hip/hip_bf16.h>
#include <math.h>

typedef __attribute__((ext_vector_type(2))) float v2f;
typedef __attribute__((ext_vector_type(8))) float v8f;

#define MM_EPS 1e-4f

__device__ __forceinline__ float leakyf(float x) { return x >= 0.f ? x : 0.01f * x; }

// ---------------------------------------------------------------------------
// Cofactor-based 4x4 inverse (row-major), matches jnp.linalg.inv numerics well
// ---------------------------------------------------------------------------
__device__ void inv4(const float* a, float* o) {
    float A2323 = a[10]*a[15] - a[11]*a[14];
    float A1323 = a[ 9]*a[15] - a[11]*a[13];
    float A1223 = a[ 9]*a[14] - a[10]*a[13];
    float A0323 = a[ 8]*a[15] - a[11]*a[12];
    float A0223 = a[ 8]*a[14] - a[10]*a[12];
    float A0123 = a[ 8]*a[13] - a[ 9]*a[12];
    float A2313 = a[ 6]*a[15] - a[ 7]*a[14];
    float A1313 = a[ 5]*a[15] - a[ 7]*a[13];
    float A1213 = a[ 5]*a[14] - a[ 6]*a[13];
    float A2312 = a[ 6]*a[11] - a[ 7]*a[10];
    float A1312 = a[ 5]*a[11] - a[ 7]*a[ 9];
    float A1212 = a[ 5]*a[10] - a[ 6]*a[ 9];
    float A0313 = a[ 4]*a[15] - a[ 7]*a[12];
    float A0213 = a[ 4]*a[14] - a[ 6]*a[12];
    float A0312 = a[ 4]*a[11] - a[ 7]*a[ 8];
    float A0212 = a[ 4]*a[10] - a[ 6]*a[ 8];
    float A0113 = a[ 4]*a[13] - a[ 5]*a[12];
    float A0112 = a[ 4]*a[ 9] - a[ 5]*a[ 8];

    float det = a[0]*(a[5]*A2323 - a[6]*A1323 + a[7]*A1223)
              - a[1]*(a[4]*A2323 - a[6]*A0323 + a[7]*A0223)
              + a[2]*(a[4]*A1323 - a[5]*A0323 + a[7]*A0123)
              - a[3]*(a[4]*A1223 - a[5]*A0223 + a[6]*A0123);
    float r = 1.0f / det;

    o[ 0] =  r*(a[5]*A2323 - a[6]*A1323 + a[7]*A1223);
    o[ 1] = -r*(a[1]*A2323 - a[2]*A1323 + a[3]*A1223);
    o[ 2] =  r*(a[1]*A2313 - a[2]*A1313 + a[3]*A1213);
    o[ 3] = -r*(a[1]*A2312 - a[2]*A1312 + a[3]*A1212);
    o[ 4] = -r*(a[4]*A2323 - a[6]*A0323 + a[7]*A0223);
    o[ 5] =  r*(a[0]*A2323 - a[2]*A0323 + a[3]*A0223);
    o[ 6] = -r*(a[0]*A2313 - a[2]*A0313 + a[3]*A0213);
    o[ 7] =  r*(a[0]*A2312 - a[2]*A0312 + a[3]*A0212);
    o[ 8] =  r*(a[4]*A1323 - a[5]*A0323 + a[7]*A0123);
    o[ 9] = -r*(a[0]*A1323 - a[1]*A0323 + a[3]*A0123);
    o[10] =  r*(a[0]*A1313 - a[1]*A0313 + a[3]*A0113);
    o[11] = -r*(a[0]*A1312 - a[1]*A0312 + a[3]*A0112);
    o[12] = -r*(a[4]*A1223 - a[5]*A0223 + a[6]*A0123);
    o[13] =  r*(a[0]*A1223 - a[1]*A0223 + a[2]*A0123);
    o[14] = -r*(a[0]*A1213 - a[1]*A0213 + a[2]*A0113);
    o[15] =  r*(a[0]*A1212 - a[1]*A0212 + a[2]*A0112);
}

// ---------------------------------------------------------------------------
// Weight repack kernels: OIHW -> [Ktot/2][Npad][2] so a lane's B fragment is a
// contiguous float2 (k = (dy*3+dx)*CK + cin, zero-padded for the 17ch conv).
// ---------------------------------------------------------------------------
__global__ void repack48(const float* __restrict__ w, float* __restrict__ wr) {
    int i = blockIdx.x * 256 + threadIdx.x;              // over 432*48
    if (i >= 432 * 48) return;
    int n = i % 48, k = i / 48;
    int tap = k / 48, c = k % 48, dy = tap / 3, dx = tap % 3;
    wr[((k >> 1) * 48 + n) * 2 + (k & 1)] = w[((n * 48 + c) * 3 + dy) * 3 + dx];
}

__global__ void repack17(const float* __restrict__ w, float* __restrict__ wr) {
    int i = blockIdx.x * 256 + threadIdx.x;              // over 180*32 (CK=20, Np=32)
    if (i >= 180 * 32) return;
    int n = i % 32, k = i / 32;
    int tap = k / 20, c = k % 20, dy = tap / 3, dx = tap % 3;
    float v = 0.f;
    if (n < 17 && c < 17) v = w[((n * 17 + c) * 3 + dy) * 3 + dx];
    wr[((k >> 1) * 32 + n) * 2 + (k & 1)] = v;
}

// ---------------------------------------------------------------------------
// 3x3 SAME conv (+bias+leaky) as implicit GEMM with V_WMMA_F32_16X16X4_F32.
// Block = 256 threads (8 waves). Each block: one row y, a 128-pixel strip.
// LDS patch layout: [3][130][CSTR] floats, channel-innermost (CSTR padded for
// conflict-free 8B-aligned ds reads). Wave w owns pixels [16w,16w+16).
// A-frag (16x4 f32): lanes 0-15 = K{0,1} (.x/.y), lanes 16-31 = K{2,3}.
// C/D (16x16 f32):   vgpr v: lanes 0-15 M=v, lanes 16-31 M=v+8; N=lane&15.
// ---------------------------------------------------------------------------
template <int CIN, int CK, int CSTR, int NT, int NOUT>
__global__ __launch_bounds__(256) void conv3x3_wmma(
    const float* __restrict__ in, const float* __restrict__ wr,
    const float* __restrict__ bias, float* __restrict__ out,
    int H, int W, int outC, int outCoff)
{
    const int NP = NT * 16;
    int strips = W >> 7;
    int bid = blockIdx.x;
    int strip = bid % strips; bid /= strips;
    int y = bid % H;
    int b = bid / H;
    int x0 = strip << 7;

    __shared__ float patch[3 * 130 * CSTR];

    int tid = threadIdx.x;
    long cs = (long)H * W;
    const float* inb = in + (long)b * CIN * cs;
    // zero-padded halo fill ('SAME' => zero padding); global reads coalesced in x
    for (int i = tid; i < 3 * 130 * CK; i += 256) {
        int dy = i / (130 * CK);
        int r  = i - dy * (130 * CK);
        int c  = r / 130;
        int xx = r - c * 130;
        int gy = y + dy - 1;
        int gx = x0 + xx - 1;
        float v = 0.f;
        if (c < CIN && gy >= 0 && gy < H && gx >= 0 && gx < W)
            v = inb[(long)c * cs + (long)gy * W + gx];
        patch[(dy * 130 + xx) * CSTR + c] = v;
    }
    __syncthreads();

    int lane = tid & 31;
    int wv   = tid >> 5;          // x-subtile 0..7
    int mrow = lane & 15;         // M for A fragment
    int half = lane >> 4;
    int ncol = lane & 15;         // N for B/D fragments

    v8f acc[NT];
#pragma unroll
    for (int t = 0; t < NT; ++t)
#pragma unroll
        for (int j = 0; j < 8; ++j) acc[t][j] = 0.f;

#pragma unroll
    for (int dy = 0; dy < 3; ++dy)
#pragma unroll
        for (int dx = 0; dx < 3; ++dx) {
            int rowbase = (dy * 130 + wv * 16 + mrow + dx) * CSTR;
            int kb0 = (dy * 3 + dx) * CK;
            for (int cb = 0; cb < CK; cb += 4) {
                v2f a;
                int ao = rowbase + cb + 2 * half;   // even dword -> b64-aligned
                a.x = patch[ao];
                a.y = patch[ao + 1];
                int kp = ((kb0 + cb) >> 1) + half;
#pragma unroll
                for (int t = 0; t < NT; ++t) {
                    const float* wp = wr + ((long)kp * NP + t * 16 + ncol) * 2;
                    v2f bb;
                    bb.x = wp[0];
                    bb.y = wp[1];
                    acc[t] = __builtin_amdgcn_wmma_f32_16x16x4_f32(
                        false, a, false, bb, (short)0, acc[t], false, false);
                }
            }
        }

#pragma unroll
    for (int t = 0; t < NT; ++t) {
        int n = t * 16 + ncol;
        if (n < NOUT) {
            float bv = bias[n];
            float* ob = out + ((long)b * outC + outCoff + n) * cs + (long)y * W + x0 + wv * 16;
#pragma unroll
            for (int v = 0; v < 8; ++v) {
                int pix = v + 8 * half;
                ob[pix] = leakyf(acc[t][v] + bv);
            }
        }
    }
}

// ---------------------------------------------------------------------------
// compute_mm + intensity: (2,48,H,W) -> (2,17,H,W), one thread per pixel
// ---------------------------------------------------------------------------
__global__ __launch_bounds__(256) void mm_features_k(
    const float* __restrict__ y, float* __restrict__ mo, int H, int W)
{
    long cs = (long)H * W;
    long idx = (long)blockIdx.x * 256 + threadIdx.x;
    if (idx >= 2 * cs) return;
    int b = (int)(idx / cs);
    long p = idx % cs;
    const float* base = y + (long)b * 48 * cs + p;

    float F[16], A[16], Wm[16];
    float s = 0.f;
#pragma unroll
    for (int i = 0; i < 16; ++i) {
        F[i]  = base[(long)i * cs];
        s    += F[i];
        A[i]  = base[(long)(16 + i) * cs];
        Wm[i] = base[(long)(32 + i) * cs];
    }
    A[0] += MM_EPS;  A[5] += MM_EPS;  A[10] += MM_EPS;  A[15] += MM_EPS;
    Wm[0] += MM_EPS; Wm[5] += MM_EPS; Wm[10] += MM_EPS; Wm[15] += MM_EPS;

    float iA[16], iW[16], T[16], M[16];
    inv4(A, iA);
    inv4(Wm, iW);
#pragma unroll
    for (int i = 0; i < 4; ++i)
#pragma unroll
        for (int j = 0; j < 4; ++j) {
            float acc = 0.f;
#pragma unroll
            for (int k = 0; k < 4; ++k) acc += iA[i * 4 + k] * F[k * 4 + j];
            T[i * 4 + j] = acc;
        }
#pragma unroll
    for (int i = 0; i < 4; ++i)
#pragma unroll
        for (int j = 0; j < 4; ++j) {
            float acc = 0.f;
#pragma unroll
            for (int k = 0; k < 4; ++k) acc += T[i * 4 + k] * iW[k * 4 + j];
            M[i * 4 + j] = acc;
        }

    float* ob = mo + (long)b * 17 * cs + p;
    ob[0] = s * (1.f / 16.f);
#pragma unroll
    for (int i = 0; i < 16; ++i) ob[(long)(1 + i) * cs] = M[i];
}

// ---------------------------------------------------------------------------
// 4x4 maxpool: (2,48,H,W) -> (2,48,H/4,W/4)
// ---------------------------------------------------------------------------
__global__ __launch_bounds__(256) void maxpool4_k(
    const float* __restrict__ in, float* __restrict__ out, int H, int W)
{
    int Ho = H >> 2, Wo = W >> 2;
    long idx = (long)blockIdx.x * 256 + threadIdx.x;
    long total = (long)2 * 48 * Ho * Wo;
    if (idx >= total) return;
    int xo = (int)(idx % Wo);
    long r = idx / Wo;
    int yo = (int)(r % Ho); r /= Ho;
    int c  = (int)(r % 48);
    int b  = (int)(r / 48);
    const float* ib = in + (((long)b * 48 + c) * H + yo * 4) * W + xo * 4;
    float mx = -3.402823466e38f;
#pragma unroll
    for (int j = 0; j < 4; ++j)
#pragma unroll
        for (int i = 0; i < 4; ++i) mx = fmaxf(mx, ib[(long)j * W + i]);
    out[idx] = mx;
}

// ---------------------------------------------------------------------------
// Bicubic (a=-0.75, align_corners, border-replicated index gather) upsample
// weights computed in f64 like the numpy reference.
// ---------------------------------------------------------------------------
__device__ __forceinline__ double cubw(double t) {
    t = fabs(t);
    const double a = -0.75;
    if (t <= 1.0) return ((a + 2.0) * t - (a + 3.0)) * t * t + 1.0;
    if (t <  2.0) return ((a * t - 5.0 * a) * t + 8.0 * a) * t - 4.0 * a;
    return 0.0;
}

__global__ __launch_bounds__(256) void upsample_rows(
    const float* __restrict__ in, float* __restrict__ out,
    int inH, int outH, int W)
{
    long idx = (long)blockIdx.x * 256 + threadIdx.x;
    long total = (long)2 * 17 * outH * W;
    if (idx >= total) return;
    int x = (int)(idx % W);
    long r = idx / W;
    int O = (int)(r % outH); r /= outH;
    int c = (int)(r % 17);
    int b = (int)(r / 17);
    double scale = (double)(inH - 1) / (double)(outH - 1);
    double src = (double)O * scale;
    int i0 = (int)floor(src);
    const float* ib = in + ((long)b * 17 + c) * inH * W + x;
    float acc = 0.f;
#pragma unroll
    for (int k = -1; k <= 2; ++k) {
        float w = (float)cubw(src - (double)(i0 + k));
        int ii = i0 + k; ii = ii < 0 ? 0 : (ii > inH - 1 ? inH - 1 : ii);
        acc += w * ib[(long)ii * W];
    }
    out[((long)b * 17 + c) * outH * W + (long)O * W + x] = acc;
}

__global__ __launch_bounds__(256) void upsample_cols(
    const float* __restrict__ in, float* __restrict__ out,
    int inW, int outW, int H)
{
    long idx = (long)blockIdx.x * 256 + threadIdx.x;
    long total = (long)2 * 17 * H * outW;
    if (idx >= total) return;
    int P = (int)(idx % outW);
    long r = idx / outW;
    int y = (int)(r % H); r /= H;
    int c = (int)(r % 17);
    int b = (int)(r / 17);
    double scale = (double)(inW - 1) / (double)(outW - 1);
    double src = (double)P * scale;
    int j0 = (int)floor(src);
    const float* ib = in + (((long)b * 17 + c) * H + y) * inW;
    float acc = 0.f;
#pragma unroll
    for (int k = -1; k <= 2; ++k) {
        float w = (float)cubw(src - (double)(j0 + k));
        int jj = j0 + k; jj = jj < 0 ? 0 : (jj > inW - 1 ? inW - 1 : jj);
        acc += w * ib[jj];
    }
    out[(((long)b * 17 + c) * H + y) * outW + P] = acc;
}

// ---------------------------------------------------------------------------
extern "C" void kernel_launch(void* const* d_in, const int* in_sizes, int n_in,
                              void* d_out, int out_size, void* d_ws, size_t ws_size,
                              hipStream_t stream)
{
    (void)in_sizes; (void)n_in; (void)out_size; (void)ws_size;
    const float* x   = (const float*)d_in[0];
    const float* iW0 = (const float*)d_in[1];
    const float* iB0 = (const float*)d_in[2];
    const float* iW1 = (const float*)d_in[3];
    const float* iB1 = (const float*)d_in[4];
    const float* oW0 = (const float*)d_in[5];
    const float* oB0 = (const float*)d_in[6];
    const float* oW1 = (const float*)d_in[7];
    const float* oB1 = (const float*)d_in[8];
    float* out = (float*)d_out;

    char* ws = (char*)d_ws;
    size_t off = 0;
    auto carve = [&](size_t bytes) {
        size_t o = off;
        off = (off + bytes + 255) & ~(size_t)255;
        return o;
    };
    float* Y     = (float*)(ws + carve(2ull * 48 * 512 * 512 * 4));  // conv48 out (both lvls)
    float* M     = (float*)(ws + carve(2ull * 17 * 512 * 512 * 4));  // mm feats / upsampled
    float* X1    = (float*)(ws + carve(2ull * 48 * 128 * 128 * 4));  // pooled
    float* M1    = (float*)(ws + carve(2ull * 17 * 128 * 128 * 4));  // lvl1 feats
    float* MR    = (float*)(ws + carve(2ull * 17 * 512 * 128 * 4));  // row-upsampled
    float* WR48a = (float*)(ws + carve(216ull * 48 * 2 * 4));
    float* WR48b = (float*)(ws + carve(216ull * 48 * 2 * 4));
    float* WR17a = (float*)(ws + carve(90ull * 32 * 2 * 4));
    float* WR17b = (float*)(ws + carve(90ull * 32 * 2 * 4));

    // weight repacks
    repack48<<<(432 * 48 + 255) / 256, 256, 0, stream>>>(iW0, WR48a);
    repack48<<<(432 * 48 + 255) / 256, 256, 0, stream>>>(iW1, WR48b);
    repack17<<<(180 * 32 + 255) / 256, 256, 0, stream>>>(oW0, WR17a);
    repack17<<<(180 * 32 + 255) / 256, 256, 0, stream>>>(oW1, WR17b);

    // ---- level 0 (512x512) ----
    conv3x3_wmma<48, 48, 50, 3, 48><<<2 * 512 * 4, 256, 0, stream>>>(
        x, WR48a, iB0, Y, 512, 512, 48, 0);
    mm_features_k<<<(int)((2ll * 512 * 512 + 255) / 256), 256, 0, stream>>>(Y, M, 512, 512);
    // level-0 upsample is identity (512->512)
    conv3x3_wmma<17, 20, 22, 2, 17><<<2 * 512 * 4, 256, 0, stream>>>(
        M, WR17a, oB0, out, 512, 512, 34, 0);
    maxpool4_k<<<(int)((2ll * 48 * 128 * 128 + 255) / 256), 256, 0, stream>>>(Y, X1, 512, 512);

    // ---- level 1 (128x128 -> upsample to 512x512) ----
    conv3x3_wmma<48, 48, 50, 3, 48><<<2 * 128 * 1, 256, 0, stream>>>(
        X1, WR48b, iB1, Y, 128, 128, 48, 0);
    mm_features_k<<<(int)((2ll * 128 * 128 + 255) / 256), 256, 0, stream>>>(Y, M1, 128, 128);
    upsample_rows<<<(int)((2ll * 17 * 512 * 128 + 255) / 256), 256, 0, stream>>>(
        M1, MR, 128, 512, 128);
    upsample_cols<<<(int)((2ll * 17 * 512 * 512 + 255) / 256), 256, 0, stream>>>(
        MR, M, 128, 512, 512);
    conv3x3_wmma<17, 20, 22, 2, 17><<<2 * 512 * 4, 256, 0, stream>>>(
        M, WR17b, oB1, out, 512, 512, 34, 17);
}